// RoiPoolingConv_30039001268965
// MI455X (gfx1250) — compile-verified
//
#include <hip/hip_runtime.h>
#include <hip/hip_bf16.h>

#define IMG_H 200
#define IMG_W 200
#define IMG_C 512
#define POOL  7
#define NPTS  (POOL * POOL)   // 49 sample points per ROI
#define TPB   128             // one float4 of channels per lane (128*4 == 512)

typedef float v4f __attribute__((ext_vector_type(4)));

__device__ __forceinline__ unsigned lds_off_of(const void* p) {
    // Generic (flat) pointers to LDS are {SHARED_BASE[..], lds_offset[31:0]}
    // on CDNA5, so truncation yields the LDS byte offset used by async loads.
    // The ptr->int cast also escapes the shared buffer so the asm "memory"
    // clobber is treated as a potential writer (keeps ds_loads live/ordered).
    return (unsigned)(size_t)p;
}

__device__ __forceinline__ void async_gather_b128(unsigned lds_byte_off,
                                                  const float* gptr) {
    // CDNA5 async global->LDS DMA (tracked by ASYNCcnt, not LOADcnt).
    asm volatile("global_load_async_to_lds_b128 %0, %1, off"
                 :
                 : "v"(lds_byte_off), "v"((unsigned long long)(size_t)gptr)
                 : "memory");
}

__device__ __forceinline__ float lerp_ref(float a, float b, float t) {
    return a + (b - a) * t;   // matches reference: a + (b-a)*t
}

__global__ __launch_bounds__(TPB) void roi_crop_resize_kernel(
    const float* __restrict__ img,   // (1, 200, 200, 512) NHWC
    const float* __restrict__ rois,  // (1000, 5): [bi, y1, x1, y2, x2]
    float* __restrict__ out,         // (1000, 7, 7, 512)
    int nrois)
{
    __shared__ v4f lbuf[2][4][TPB];  // 2-deep ring x 4 corners x 128 float4 = 16 KB

    const int tid = threadIdx.x;
    const int n   = blockIdx.x;
    if (n >= nrois) return;

    const float* r = rois + (size_t)n * 5;
    const int   bi = (int)r[0];
    const float y1 = r[1], x1 = r[2], y2 = r[3], x2 = r[4];

    const float oy = y1 * (float)(IMG_H - 1);
    const float ox = x1 * (float)(IMG_W - 1);
    const float sy = (y2 - y1) * (float)(IMG_H - 1) / (float)(POOL - 1);
    const float sx = (x2 - x1) * (float)(IMG_W - 1) / (float)(POOL - 1);

    // Lane owns channels [4*tid, 4*tid+4)
    const float* base = img + (size_t)bi * IMG_H * IMG_W * IMG_C + (size_t)tid * 4;
    float*       outp = out + (size_t)n * NPTS * IMG_C + (size_t)tid * 4;

    auto stage = [&](int p, int buf) {
        const int py = p / POOL, px = p % POOL;
        const float ys = oy + (float)py * sy;
        const float xs = ox + (float)px * sx;
        const float yf = floorf(ys), xf = floorf(xs);
        const int y0i = min(max((int)yf,     0), IMG_H - 1);
        const int y1i = min(max((int)yf + 1, 0), IMG_H - 1);
        const int x0i = min(max((int)xf,     0), IMG_W - 1);
        const int x1i = min(max((int)xf + 1, 0), IMG_W - 1);
        const float* g0 = base + ((size_t)y0i * IMG_W + x0i) * IMG_C;
        const float* g1 = base + ((size_t)y0i * IMG_W + x1i) * IMG_C;
        const float* g2 = base + ((size_t)y1i * IMG_W + x0i) * IMG_C;
        const float* g3 = base + ((size_t)y1i * IMG_W + x1i) * IMG_C;
        async_gather_b128(lds_off_of(&lbuf[buf][0][tid]), g0);
        async_gather_b128(lds_off_of(&lbuf[buf][1][tid]), g1);
        async_gather_b128(lds_off_of(&lbuf[buf][2][tid]), g2);
        async_gather_b128(lds_off_of(&lbuf[buf][3][tid]), g3);
    };

    auto finish = [&](int p, int buf) {
        const int py = p / POOL, px = p % POOL;
        const float ys = oy + (float)py * sy;
        const float xs = ox + (float)px * sx;
        const float yf = floorf(ys), xf = floorf(xs);
        const float ly = ys - yf;
        const float lx = xs - xf;
        const bool valid = (ys >= 0.f) && (ys <= (float)(IMG_H - 1)) &&
                           (xs >= 0.f) && (xs <= (float)(IMG_W - 1));

        const v4f f00 = lbuf[buf][0][tid];   // ds_load_b128
        const v4f f01 = lbuf[buf][1][tid];
        const v4f f10 = lbuf[buf][2][tid];
        const v4f f11 = lbuf[buf][3][tid];

        v4f res;
        res.x = lerp_ref(lerp_ref(f00.x, f01.x, lx), lerp_ref(f10.x, f11.x, lx), ly);
        res.y = lerp_ref(lerp_ref(f00.y, f01.y, lx), lerp_ref(f10.y, f11.y, lx), ly);
        res.z = lerp_ref(lerp_ref(f00.z, f01.z, lx), lerp_ref(f10.z, f11.z, lx), ly);
        res.w = lerp_ref(lerp_ref(f00.w, f01.w, lx), lerp_ref(f10.w, f11.w, lx), ly);
        if (!valid) { res.x = 0.f; res.y = 0.f; res.z = 0.f; res.w = 0.f; }

        // Streaming 100 MB output: non-temporal so the 82 MB image stays in
        // the 192 MB L2 for the whole kernel.
        __builtin_nontemporal_store(res, (v4f*)(outp + (size_t)p * IMG_C));
    };

    // Software pipeline over the 49 sample points, double-buffered in LDS.
    stage(0, 0);
    for (int p = 0; p < NPTS; ++p) {
        if (p + 1 < NPTS) {
            stage(p + 1, (p + 1) & 1);
            // 8 async loads outstanding; <=4 left  ==> point p's 4 are done
            asm volatile("s_wait_asynccnt 4" ::: "memory");
        } else {
            asm volatile("s_wait_asynccnt 0" ::: "memory");
        }
        finish(p, p & 1);
    }
}

extern "C" void kernel_launch(void* const* d_in, const int* in_sizes, int n_in,
                              void* d_out, int out_size, void* d_ws, size_t ws_size,
                              hipStream_t stream) {
    const float* img  = (const float*)d_in[0];   // (1,200,200,512) f32
    const float* rois = (const float*)d_in[1];   // (1,1000,5) f32
    float* out = (float*)d_out;                  // (1,1000,7,7,512) f32
    const int nrois = in_sizes[1] / 5;           // 1000

    roi_crop_resize_kernel<<<nrois, TPB, 0, stream>>>(img, rois, out, nrois);
}